// GeluAndMul_6253472383788
// MI455X (gfx1250) — compile-verified
//
#include <hip/hip_runtime.h>

#define TOKENS 4096
#define DFF    14336

typedef float v4f __attribute__((ext_vector_type(4)));

// 3584 float4 per output row = 7 blocks x 256 threads x 2 float4/thread.
// Each thread handles col4 and col4+1792: every load/store instruction is a
// fully coalesced 32-lane b128 NT access. Unroll-by-2 amortizes the per-wave
// SALU preamble (workgroup-ID decode + base address math) over 2x the bytes.
__global__ __launch_bounds__(256) void gelu_and_mul_kernel(
    const float* __restrict__ x, float* __restrict__ out) {
  const int row   = blockIdx.y;                              // 0..4095
  const int col4a = blockIdx.x * blockDim.x + threadIdx.x;   // 0..1791
  // second element at +1792 float4 (exact second half of the row's float4s)

  const size_t grow = (size_t)row * (2 * DFF / 4);
  const size_t orow = (size_t)row * (DFF / 4);

  const v4f* gate0 = (const v4f*)x + grow + col4a;
  const v4f* gate1 = gate0 + 1792;
  const v4f* up0   = gate0 + (DFF / 4);
  const v4f* up1   = gate1 + (DFF / 4);

  v4f g0 = __builtin_nontemporal_load(gate0);
  v4f u0 = __builtin_nontemporal_load(up0);
  v4f g1 = __builtin_nontemporal_load(gate1);
  v4f u1 = __builtin_nontemporal_load(up1);

  const float c = 0.7978845608028654f;  // sqrt(2/pi)
  const float k = 0.044715f;

  v4f r0, r1;
#pragma unroll
  for (int i = 0; i < 4; ++i) {
    {
      float gv = g0[i];
      float t  = c * (gv + k * gv * gv * gv);
      float th;
#if __has_builtin(__builtin_amdgcn_tanhf)
      th = __builtin_amdgcn_tanhf(t);        // v_tanh_f32 (gfx1250 TRANS op)
#else
      asm volatile("v_tanh_f32 %0, %1\n\tv_nop" : "=v"(th) : "v"(t));
#endif
      r0[i] = 0.5f * gv * (1.0f + th) * u0[i];
    }
    {
      float gv = g1[i];
      float t  = c * (gv + k * gv * gv * gv);
      float th;
#if __has_builtin(__builtin_amdgcn_tanhf)
      th = __builtin_amdgcn_tanhf(t);
#else
      asm volatile("v_tanh_f32 %0, %1\n\tv_nop" : "=v"(th) : "v"(t));
#endif
      r1[i] = 0.5f * gv * (1.0f + th) * u1[i];
    }
  }

  v4f* o0 = (v4f*)out + orow + col4a;
  v4f* o1 = o0 + 1792;
  __builtin_nontemporal_store(r0, o0);
  __builtin_nontemporal_store(r1, o1);
}

extern "C" void kernel_launch(void* const* d_in, const int* in_sizes, int n_in,
                              void* d_out, int out_size, void* d_ws, size_t ws_size,
                              hipStream_t stream) {
  const float* x   = (const float*)d_in[0];
  float*       out = (float*)d_out;

  dim3 block(256);
  dim3 grid(7, TOKENS);  // 7*256*2 = 3584 float4 per row, exact coverage
  gelu_and_mul_kernel<<<grid, block, 0, stream>>>(x, out);
}